// EdgeGCN_DIR_CAT_52364241273343
// MI455X (gfx1250) — compile-verified
//
#include <hip/hip_runtime.h>

// ---------------------------------------------------------------------------
// EdgeGCN on MI455X (gfx1250): fp32 WMMA 16x16x4, fused reductions.
// Memory-bound problem (~780 MB @ 23.3 TB/s => ~33us floor); every byte of the
// big [1M x 32] edge tensors is read once and the projected [1M x 64] tensors
// are written once, with the axis-sums accumulated in registers.
// ---------------------------------------------------------------------------

typedef float v2f __attribute__((ext_vector_type(2)));
typedef float v8f __attribute__((ext_vector_type(8)));

__device__ __forceinline__ v8f wmma4(v2f a, v2f b, v8f c) {
  // D = A(16x4 f32) x B(4x16 f32) + C(16x16 f32)
  return __builtin_amdgcn_wmma_f32_16x16x4_f32(
      /*neg_a=*/false, a, /*neg_b=*/false, b,
      /*c_mod=*/(short)0, c, /*reuse_a=*/false, /*reuse_b=*/false);
}

__device__ __forceinline__ v2f ldg2(const float* p) { return *(const v2f*)p; }

// ---------------------------------------------------------------------------
// Edge projection: out_m[R, o] = sum_e feat[R, e] * W[e, o], R = n*1024+m
// MODE 0 (edge_in):  also sum over m  -> cat[n,   64 + o]
// MODE 1 (edge_out): also sum over n  -> cat[m,  128 + o]
// grid = 2048 blocks x 128 threads = 8192 waves; each wave: 8 strips of 16 rows
// ---------------------------------------------------------------------------
template <int MODE>
__global__ __launch_bounds__(128) void proj_kernel(
    const float* __restrict__ feat,   // [1024*1024, 32]
    const float* __restrict__ W,      // [32, 64]
    float* __restrict__ out_m,        // [1024*1024, 64]
    float* __restrict__ cat)          // [1024, 192] (atomic accum region)
{
  __shared__ float bt[64 * 32];       // Bt[o][k] = W[k][o]
  const int tid = threadIdx.x;
  for (int i = tid; i < 64 * 32; i += 128)
    bt[i] = W[(i & 31) * 64 + (i >> 5)];
  __syncthreads();

  const int lane = tid & 31;
  const int nn = lane & 15;           // column / row-in-strip index
  const int hi = lane >> 4;           // half-wave select
  const int kb = hi * 2;              // K sub-offset for A/B frags

  const int wid = blockIdx.x * 4 + (tid >> 5);   // 0..8191

  int n_fix, mt_fix;
  if (MODE == 0) { n_fix = wid >> 3;  mt_fix = wid & 7;   }
  else           { mt_fix = wid >> 7; n_fix  = wid & 127; }

  float csum[4] = {0.f, 0.f, 0.f, 0.f};  // MODE 0: per-column partial sums
  v8f   ssum[4] = {};                    // MODE 1: full-tile partial sums

  for (int it = 0; it < 8; ++it) {
    const int n  = (MODE == 0) ? n_fix : (n_fix + it * 128);
    const int mt = (MODE == 0) ? (mt_fix + it * 8) : mt_fix;
    const long rowbase = (long)n * 1024 + (long)mt * 16;

    // A strip 16x32: lane holds row (rowbase+nn), K subset per half-wave
    v2f a[8];
    const float* ap = feat + (rowbase + nn) * 32 + kb;
    #pragma unroll
    for (int k0 = 0; k0 < 8; ++k0) a[k0] = ldg2(ap + k0 * 4);

    #pragma unroll
    for (int t = 0; t < 4; ++t) {
      v8f acc = {};
      const float* bp = bt + (t * 16 + nn) * 32 + kb;
      #pragma unroll
      for (int k0 = 0; k0 < 8; ++k0) {
        v2f b = *(const v2f*)(bp + k0 * 4);
        acc = wmma4(a[k0], b, acc);
      }
      // store 16x16 tile (VGPR r -> row rowbase + hi*8 + r, col t*16+nn)
      float* op = out_m + (rowbase + hi * 8) * 64 + t * 16 + nn;
      #pragma unroll
      for (int r = 0; r < 8; ++r) op[(long)r * 64] = acc[r];

      if (MODE == 0) {
        float s = 0.f;
        #pragma unroll
        for (int r = 0; r < 8; ++r) s += acc[r];
        csum[t] += s;
      } else {
        ssum[t] += acc;
      }
    }
  }

  if (MODE == 0) {
    #pragma unroll
    for (int t = 0; t < 4; ++t) {
      float s = csum[t] + __shfl_down(csum[t], 16, 32);
      if (lane < 16)
        atomicAdd(&cat[(long)n_fix * 192 + 64 + t * 16 + nn], s);
    }
  } else {
    const int m0 = mt_fix * 16 + hi * 8;
    #pragma unroll
    for (int t = 0; t < 4; ++t)
      #pragma unroll
      for (int r = 0; r < 8; ++r)
        atomicAdd(&cat[(long)(m0 + r) * 192 + 128 + t * 16 + nn], ssum[t][r]);
  }
}

// ---------------------------------------------------------------------------
// support = x[1024,128] @ W_node[128,32]  (both in & out weights)
// 16 blocks x 128 threads = 64 waves, one 16-row strip each
// ---------------------------------------------------------------------------
__global__ __launch_bounds__(128) void support_kernel(
    const float* __restrict__ x, const float* __restrict__ Wni,
    const float* __restrict__ Wno, float* __restrict__ sup_in,
    float* __restrict__ sup_out)
{
  const int tid = threadIdx.x;
  const int lane = tid & 31;
  const int wid = blockIdx.x * 4 + (tid >> 5);   // 0..63
  const int nn = lane & 15, hi = lane >> 4, kb = hi * 2;
  const int m0 = wid * 16;

  #pragma unroll
  for (int w = 0; w < 2; ++w) {
    const float* Wp = (w == 0) ? Wni : Wno;
    float* sup = (w == 0) ? sup_in : sup_out;
    #pragma unroll
    for (int t = 0; t < 2; ++t) {
      v8f acc = {};
      for (int k0 = 0; k0 < 128; k0 += 4) {
        v2f a = ldg2(x + (long)(m0 + nn) * 128 + k0 + kb);
        v2f b;
        b.x = Wp[(k0 + kb) * 32 + t * 16 + nn];
        b.y = Wp[(k0 + kb + 1) * 32 + t * 16 + nn];
        acc = wmma4(a, b, acc);
      }
      float* op = sup + (long)(m0 + hi * 8) * 32 + t * 16 + nn;
      #pragma unroll
      for (int r = 0; r < 8; ++r) op[r * 32] = acc[r];
    }
  }
}

// ---------------------------------------------------------------------------
// node_in  = adj^T @ sup_in  -> cat cols [0,32)
// node_out = adj   @ sup_out -> cat cols [32,64)
// ---------------------------------------------------------------------------
__global__ __launch_bounds__(128) void node_kernel(
    const float* __restrict__ adj, const float* __restrict__ sup_in,
    const float* __restrict__ sup_out, float* __restrict__ cat)
{
  const int tid = threadIdx.x;
  const int lane = tid & 31;
  const int wid = blockIdx.x * 4 + (tid >> 5);   // 0..63
  const int nn = lane & 15, hi = lane >> 4, kb = hi * 2;
  const int i0 = wid * 16;

  #pragma unroll
  for (int v = 0; v < 2; ++v) {
    const float* sup = (v == 0) ? sup_in : sup_out;
    v8f acc0 = {}, acc1 = {};
    for (int k0 = 0; k0 < 1024; k0 += 4) {
      v2f a;
      if (v == 0) {  // adj^T: A[i,k] = adj[k,i] (coalesced across lanes 0-15)
        a.x = adj[(long)(k0 + kb) * 1024 + i0 + nn];
        a.y = adj[(long)(k0 + kb + 1) * 1024 + i0 + nn];
      } else {       // adj rows, contiguous
        a = ldg2(adj + (long)(i0 + nn) * 1024 + k0 + kb);
      }
      v2f b0, b1;
      b0.x = sup[(k0 + kb) * 32 + nn];      b0.y = sup[(k0 + kb + 1) * 32 + nn];
      b1.x = sup[(k0 + kb) * 32 + 16 + nn]; b1.y = sup[(k0 + kb + 1) * 32 + 16 + nn];
      acc0 = wmma4(a, b0, acc0);
      acc1 = wmma4(a, b1, acc1);
    }
    float* op = cat + (long)(i0 + hi * 8) * 192 + v * 32;
    #pragma unroll
    for (int r = 0; r < 8; ++r) {
      op[(long)r * 192 + nn]      = acc0[r];
      op[(long)r * 192 + 16 + nn] = acc1[r];
    }
  }
}

// ---------------------------------------------------------------------------
// output = relu(cat[1024,192] @ W_agg[192,64] + bias)
// ---------------------------------------------------------------------------
__global__ __launch_bounds__(128) void final_kernel(
    const float* __restrict__ cat, const float* __restrict__ Wagg,
    const float* __restrict__ bias, float* __restrict__ out)
{
  const int tid = threadIdx.x;
  const int lane = tid & 31;
  const int wid = blockIdx.x * 4 + (tid >> 5);   // 0..63
  const int nn = lane & 15, hi = lane >> 4, kb = hi * 2;
  const int i0 = wid * 16;

  v8f acc[4] = {};
  for (int k0 = 0; k0 < 192; k0 += 4) {
    v2f a = ldg2(cat + (long)(i0 + nn) * 192 + k0 + kb);
    #pragma unroll
    for (int t = 0; t < 4; ++t) {
      v2f b;
      b.x = Wagg[(k0 + kb) * 64 + t * 16 + nn];
      b.y = Wagg[(k0 + kb + 1) * 64 + t * 16 + nn];
      acc[t] = wmma4(a, b, acc[t]);
    }
  }
  #pragma unroll
  for (int t = 0; t < 4; ++t) {
    float bsc = bias[t * 16 + nn];
    float* op = out + (long)(i0 + hi * 8) * 64 + t * 16 + nn;
    #pragma unroll
    for (int r = 0; r < 8; ++r) {
      float vv = acc[t][r] + bsc;
      op[(long)r * 64] = vv > 0.f ? vv : 0.f;
    }
  }
}

// ---------------------------------------------------------------------------
extern "C" void kernel_launch(void* const* d_in, const int* in_sizes, int n_in,
                              void* d_out, int out_size, void* d_ws, size_t ws_size,
                              hipStream_t stream) {
  (void)in_sizes; (void)n_in; (void)out_size; (void)ws_size;
  const float* x    = (const float*)d_in[0];
  const float* adj  = (const float*)d_in[1];
  const float* fin  = (const float*)d_in[2];
  const float* fout = (const float*)d_in[3];
  const float* Wni  = (const float*)d_in[4];
  const float* Wno  = (const float*)d_in[5];
  const float* Wei  = (const float*)d_in[6];
  const float* Weo  = (const float*)d_in[7];
  const float* Wagg = (const float*)d_in[8];
  const float* bias = (const float*)d_in[9];

  float* out       = (float*)d_out;                       // [1024,64]
  float* out_m_in  = out + 1024 * 64;                     // [1024,1024,64]
  float* out_m_out = out_m_in + (size_t)1024 * 1024 * 64; // [1024,1024,64]

  float* cat     = (float*)d_ws;                          // [1024,192]
  float* sup_in  = cat + 1024 * 192;                      // [1024,32]
  float* sup_out = sup_in + 1024 * 32;                    // [1024,32]

  hipMemsetAsync(cat, 0, (size_t)1024 * 192 * sizeof(float), stream);

  proj_kernel<0><<<2048, 128, 0, stream>>>(fin, Wei, out_m_in, cat);
  proj_kernel<1><<<2048, 128, 0, stream>>>(fout, Weo, out_m_out, cat);
  support_kernel<<<16, 128, 0, stream>>>(x, Wni, Wno, sup_in, sup_out);
  node_kernel<<<16, 128, 0, stream>>>(adj, sup_in, sup_out, cat);
  final_kernel<<<16, 128, 0, stream>>>(cat, Wagg, bias, out);
}